// OTLoss_9122510537223
// MI455X (gfx1250) — compile-verified
//
#include <hip/hip_runtime.h>
#include <hip/hip_bf16.h>
#include <math.h>

typedef __attribute__((ext_vector_type(2))) float v2f;
typedef __attribute__((ext_vector_type(8))) float v8f;

#define NPTS   4096
#define DIM    256
#define NEGV   (-1.0e30f)
#define BLUR2  0.0025f      /* blur^p = 0.05^2 */
#define SCAL2  0.81f        /* scaling^p = 0.9^2 */
#define NSTEPS 64

// ---------------------------------------------------------------- scalars
__global__ void ot_init_scalars(unsigned int* maxc, int* cnt) {
    if (threadIdx.x == 0) { *maxc = 0u; *cnt = 0; }
}

// ---------------------------------------------------------------- sq norms
__global__ __launch_bounds__(256) void ot_sqnorm(const float* __restrict__ x,
                                                 float* __restrict__ sq) {
    __shared__ float s[256];
    int row = blockIdx.x;
    float v = x[(size_t)row * DIM + threadIdx.x];
    s[threadIdx.x] = v * v;
    __syncthreads();
    for (int off = 128; off > 0; off >>= 1) {
        if (threadIdx.x < off) s[threadIdx.x] += s[threadIdx.x + off];
        __syncthreads();
    }
    if (threadIdx.x == 0) sq[row] = s[0];
}

// ---------------------------------------------------------------- source count
__global__ __launch_bounds__(256) void ot_count(const int* __restrict__ domain,
                                                int* cnt) {
    __shared__ int s[256];
    int i = blockIdx.x * 256 + threadIdx.x;
    s[threadIdx.x] = (domain[i] == 0) ? 1 : 0;
    __syncthreads();
    for (int off = 128; off > 0; off >>= 1) {
        if (threadIdx.x < off) s[threadIdx.x] += s[threadIdx.x + off];
        __syncthreads();
    }
    if (threadIdx.x == 0) atomicAdd(cnt, s[0]);
}

// ---------------------------------------------------------------- weights + init h
__global__ __launch_bounds__(256) void ot_prep(const int* __restrict__ domain,
                                               const int* __restrict__ cnt,
                                               float* a_log, float* a_w,
                                               float* h1, float* h2) {
    int i = blockIdx.x * 256 + threadIdx.x;
    int ns = *cnt; if (ns < 1) ns = 1;
    float fns = (float)ns;
    bool m = (domain[i] == 0);
    float al = m ? -__logf(fns) : NEGV;
    a_log[i] = al;
    a_w[i]   = m ? (1.0f / fns) : 0.0f;
    h1[i] = al;   // init softmin uses h = b_log (== a_log)
    h2[i] = al;
}

// ---------------------------------------------------------------- C = 0.5||xi-xj||^2 via WMMA
// One 16x16 tile per wave; K accumulated in chunks of 4 with V_WMMA_F32_16X16X4_F32.
__global__ __launch_bounds__(256) void ot_build_C(const float* __restrict__ x,
                                                  const float* __restrict__ sq,
                                                  float* __restrict__ C,
                                                  unsigned int* maxc) {
    const int wave = threadIdx.x >> 5;
    const int lane = threadIdx.x & 31;
    const int tile = blockIdx.x * 8 + wave;           // 256x256 tiles, 8 waves/block
    const int ti = tile >> 8;
    const int tj = tile & 255;
    const int n  = lane & 15;
    const int kk = lane >> 4;

    const float* arow = x + (size_t)(ti * 16 + n) * DIM;   // A: lanes 0-15 M=0..15 (K0,K1), 16-31 (K2,K3)
    const float* brow = x + (size_t)(tj * 16 + n) * DIM;   // B mirrors A (C symmetric anyway)

    v8f acc = {};
#if __has_builtin(__builtin_amdgcn_wmma_f32_16x16x4_f32)
    for (int k = 0; k < DIM; k += 4) {
        v2f a = *(const v2f*)(arow + k + 2 * kk);
        v2f b = *(const v2f*)(brow + k + 2 * kk);
        acc = __builtin_amdgcn_wmma_f32_16x16x4_f32(
            false, a, false, b, (short)0, acc, false, false);
    }
#else
    // scalar fallback: each lane computes its 8 C/D elements directly
    for (int v = 0; v < 8; ++v) {
        const float* ar = x + (size_t)(ti * 16 + v + 8 * kk) * DIM;
        const float* br = x + (size_t)(tj * 16 + n) * DIM;
        float d = 0.f;
        for (int k = 0; k < DIM; ++k) d = __builtin_fmaf(ar[k], br[k], d);
        acc[v] = d;
    }
#endif

    // C/D layout: VGPR v, lanes 0-15 -> (M=v, N=lane); lanes 16-31 -> (M=v+8, N=lane-16)
    float lmax = 0.f;
    const int col = tj * 16 + n;
    const float sqc = sq[col];
    for (int v = 0; v < 8; ++v) {
        int row = ti * 16 + v + 8 * kk;
        float cc = fmaxf(0.f, 0.5f * (sq[row] + sqc) - acc[v]);
        C[(size_t)row * NPTS + col] = cc;
        lmax = fmaxf(lmax, cc);
    }
    for (int off = 16; off > 0; off >>= 1)
        lmax = fmaxf(lmax, __shfl_xor(lmax, off, 32));
    if (lane == 0) atomicMax(maxc, __float_as_uint(lmax));   // C>=0 so uint-bit order == float order
}

// ---------------------------------------------------------------- eps schedule (device side: eps0 is data-dependent)
__global__ void ot_build_eps(const unsigned int* maxc, float* eps) {
    if (threadIdx.x == 0) {
        float m  = __uint_as_float(*maxc);
        float e0 = fmaxf(2.0f * m, BLUR2);
        eps[0] = e0;                         // init
        float e = e0;
        for (int s = 0; s < NSTEPS; ++s) {   // eps_seq[s] = max(e0 * 0.81^s, blur^2)
            eps[1 + s] = fmaxf(e, BLUR2);
            e *= SCAL2;
        }
        eps[65] = BLUR2;                     // final extrapolation eps
    }
}

// ---------------------------------------------------------------- fused symmetric Sinkhorn step
// One block per row i. Two tracks (f==g and da==db by symmetry).
// phase 0: init  f=da=-eps*lse       phase 1: averaged update
// phase 2: final extrapolation; writes 2*a_w*(f_fin-da_fin) into f[] for reduction.
__global__ __launch_bounds__(256) void ot_softmin_step(
    const float* __restrict__ C,
    const float* __restrict__ h1, const float* __restrict__ h2,
    float* __restrict__ f, float* __restrict__ da,
    float* __restrict__ h1n, float* __restrict__ h2n,
    const float* __restrict__ eps_arr, int eps_idx, int next_idx, int phase,
    const float* __restrict__ a_log, const float* __restrict__ a_w) {

    __shared__ float lh1[NPTS], lh2[NPTS];          // 32 KB staged h-vectors
    __shared__ float sm1[256], ss1[256], sm2[256], ss2[256];

    const int tid = threadIdx.x;
    const int row = blockIdx.x;
    const float eps = eps_arr[eps_idx];
    const float ie  = 1.0f / eps;
    const float* Crow = C + (size_t)row * NPTS;

    for (int u = tid; u < NPTS; u += 256) { lh1[u] = h1[u]; lh2[u] = h2[u]; }
    __syncthreads();

    float t1[16], t2[16];
    float m1 = -3.0e38f, m2 = -3.0e38f;
#pragma unroll
    for (int u = 0; u < 16; ++u) {
        int c = u * 256 + tid;
        float cv = Crow[c];
        float v1 = __builtin_fmaf(-cv, ie, lh1[c]);
        float v2 = __builtin_fmaf(-cv, ie, lh2[c]);
        t1[u] = v1; t2[u] = v2;
        m1 = fmaxf(m1, v1); m2 = fmaxf(m2, v2);
    }
    float s1 = 0.f, s2 = 0.f;
#pragma unroll
    for (int u = 0; u < 16; ++u) {
        s1 += __expf(t1[u] - m1);
        s2 += __expf(t2[u] - m2);
    }
    sm1[tid] = m1; ss1[tid] = s1; sm2[tid] = m2; ss2[tid] = s2;
    __syncthreads();
    for (int off = 128; off > 0; off >>= 1) {
        if (tid < off) {
            float ma = sm1[tid], mb = sm1[tid + off];
            float M  = fmaxf(ma, mb);
            ss1[tid] = ss1[tid] * __expf(ma - M) + ss1[tid + off] * __expf(mb - M);
            sm1[tid] = M;
            ma = sm2[tid]; mb = sm2[tid + off];
            M  = fmaxf(ma, mb);
            ss2[tid] = ss2[tid] * __expf(ma - M) + ss2[tid + off] * __expf(mb - M);
            sm2[tid] = M;
        }
        __syncthreads();
    }
    if (tid == 0) {
        float ft  = -eps * (sm1[0] + __logf(ss1[0]));
        float dat = -eps * (sm2[0] + __logf(ss2[0]));
        if (phase == 2) {
            f[row] = 2.0f * a_w[row] * (ft - dat);   // loss contribution, reduced later
        } else {
            float fn = (phase == 0) ? ft  : 0.5f * (f[row]  + ft);
            float dn = (phase == 0) ? dat : 0.5f * (da[row] + dat);
            f[row] = fn; da[row] = dn;
            float ien = 1.0f / eps_arr[next_idx];
            float al  = a_log[row];
            h1n[row] = al + fn * ien;                // h for NEXT step's eps
            h2n[row] = al + dn * ien;
        }
    }
}

// ---------------------------------------------------------------- deterministic loss reduction
__global__ __launch_bounds__(256) void ot_reduce(const float* __restrict__ v,
                                                 float* out) {
    __shared__ float s[256];
    float a = 0.f;
    for (int u = threadIdx.x; u < NPTS; u += 256) a += v[u];
    s[threadIdx.x] = a;
    __syncthreads();
    for (int off = 128; off > 0; off >>= 1) {
        if (threadIdx.x < off) s[threadIdx.x] += s[threadIdx.x + off];
        __syncthreads();
    }
    if (threadIdx.x == 0) out[0] = s[0];
}

// ================================================================ launch
extern "C" void kernel_launch(void* const* d_in, const int* in_sizes, int n_in,
                              void* d_out, int out_size, void* d_ws, size_t ws_size,
                              hipStream_t stream) {
    const float* x      = (const float*)d_in[0];   // (4096, 256) f32
    const int*   domain = (const int*)d_in[1];     // (4096,)     i32
    float*       out    = (float*)d_out;           // scalar f32

    float* ws = (float*)d_ws;
    float* C     = ws;                                  // 64 MB
    float* sq    = ws + (size_t)NPTS * NPTS;
    float* a_log = sq    + NPTS;
    float* a_w   = a_log + NPTS;
    float* f     = a_w   + NPTS;
    float* da    = f     + NPTS;
    float* hA1   = da    + NPTS;
    float* hA2   = hA1   + NPTS;
    float* hB1   = hA2   + NPTS;
    float* hB2   = hB1   + NPTS;
    float* eps   = hB2   + NPTS;                        // 66 entries
    unsigned int* maxc = (unsigned int*)(eps + 66);
    int*          cnt  = (int*)(eps + 67);

    ot_init_scalars<<<1, 1, 0, stream>>>(maxc, cnt);
    ot_sqnorm<<<NPTS, 256, 0, stream>>>(x, sq);
    ot_count<<<NPTS / 256, 256, 0, stream>>>(domain, cnt);
    ot_prep<<<NPTS / 256, 256, 0, stream>>>(domain, cnt, a_log, a_w, hA1, hA2);
    ot_build_C<<<(NPTS / 16) * (NPTS / 16) / 8, 256, 0, stream>>>(x, sq, C, maxc);
    ot_build_eps<<<1, 1, 0, stream>>>(maxc, eps);

    // init potentials at eps0 (phase 0): reads hA, writes hB for step 0
    ot_softmin_step<<<NPTS, 256, 0, stream>>>(C, hA1, hA2, f, da, hB1, hB2,
                                              eps, 0, 1, 0, a_log, a_w);
    int cur = 1;  // 1 -> current h in B buffers
    for (int s = 0; s < NSTEPS; ++s) {
        float* hc1 = cur ? hB1 : hA1; float* hc2 = cur ? hB2 : hA2;
        float* hn1 = cur ? hA1 : hB1; float* hn2 = cur ? hA2 : hB2;
        int nxt = (s == NSTEPS - 1) ? 65 : (s + 2);
        ot_softmin_step<<<NPTS, 256, 0, stream>>>(C, hc1, hc2, f, da, hn1, hn2,
                                                  eps, s + 1, nxt, 1, a_log, a_w);
        cur ^= 1;
    }
    // final extrapolation at eps = blur^2 (phase 2): contributions into f[]
    float* hf1 = cur ? hB1 : hA1; float* hf2 = cur ? hB2 : hA2;
    ot_softmin_step<<<NPTS, 256, 0, stream>>>(C, hf1, hf2, f, da, hf1, hf2,
                                              eps, 65, 65, 2, a_log, a_w);
    ot_reduce<<<1, 256, 0, stream>>>(f, out);
}